// BahdanauCrossAttention_78804059947112
// MI455X (gfx1250) — compile-verified
//
#include <hip/hip_runtime.h>
#include <hip/hip_bf16.h>

// ---------------------------------------------------------------------------
// Problem constants (B=4, NQ=128, NKV=1024, Q_DIM=512, KV_DIM=768, ATTN_DIM=256)
// ---------------------------------------------------------------------------
#define BATCH    4
#define NQ       128
#define NKV      1024
#define QDIM     512
#define KVDIM    768
#define ADIM     256

typedef __attribute__((ext_vector_type(2))) float v2f;
typedef __attribute__((ext_vector_type(4))) float v4f;
typedef __attribute__((ext_vector_type(8))) float v8f;

#define WMMA_F32(a, b, c) \
    __builtin_amdgcn_wmma_f32_16x16x4_f32(false, (a), false, (b), (short)0, (c), false, false)

// ---------------------------------------------------------------------------
// tanh: gfx1250 V_TANH_F32 (TRANS) when available, exp2-based fallback.
// ---------------------------------------------------------------------------
__device__ __forceinline__ float fast_tanh(float x) {
#if __has_builtin(__builtin_amdgcn_tanhf)
    return __builtin_amdgcn_tanhf(x);
#else
    float e = __builtin_amdgcn_exp2f(x * 2.8853900817779268f);
    return 1.0f - 2.0f * __builtin_amdgcn_rcpf(e + 1.0f);
#endif
}

// ---------------------------------------------------------------------------
// NT GEMM via V_WMMA_F32_16X16X4_F32:  C[M,N] = A[M,K] * W[N,K]^T (+ bias)
// block = 128 threads (4 waves). Wave: 32 (M) x 64 (N), 8 accumulators.
// K consumed in chunks of 8 via b128 loads; fragments for the two k-steps of
// a chunk are the .lo/.hi halves (A and B share the slot->k permutation, so
// the contraction is exact). Register double-buffering overlaps loads with
// the WMMA pipe. grid = (M/32, N/256). Requires K%8==0, lda/ldw%4==0.
//
// Fragment mapping (ISA 7.12.2, wave32), lane l (i=l&15, h=l>>4):
//   A 16x4: {v0,v1} = A[i][k+2h], A[i][k+2h+1]
//   B 4x16: {v0,v1} = B[k+2h][i], B[k+2h+1][i]  (NT: row i of W)
//   C:      VGPR v -> C[v + 8h][i]
// ---------------------------------------------------------------------------
struct FragNT { v4f a0, a1, b0, b1, b2, b3; };

__device__ __forceinline__ void load_nt(FragNT& f,
    const float* arow0, const float* arow1,
    const float* wr0, const float* wr1, const float* wr2, const float* wr3, int ko)
{
    f.a0 = *(const v4f*)(arow0 + ko);
    f.a1 = *(const v4f*)(arow1 + ko);
    f.b0 = *(const v4f*)(wr0 + ko);
    f.b1 = *(const v4f*)(wr1 + ko);
    f.b2 = *(const v4f*)(wr2 + ko);
    f.b3 = *(const v4f*)(wr3 + ko);
}

__global__ __launch_bounds__(128) void gemm_nt_wmma(
    const float* __restrict__ A, const float* __restrict__ W,
    const float* __restrict__ bias, float* __restrict__ C,
    int K, int lda, int ldw, int ldc)
{
    const int lane = threadIdx.x & 31;
    const int wave = threadIdx.x >> 5;
    const int m0   = blockIdx.x * 32;
    const int n0   = blockIdx.y * 256 + wave * 64;
    const int i    = lane & 15;
    const int h    = lane >> 4;

    const float* arow0 = A + (size_t)(m0 + i) * lda;
    const float* arow1 = arow0 + (size_t)16 * lda;
    const float* wr0   = W + (size_t)(n0 + i +  0) * ldw;
    const float* wr1   = W + (size_t)(n0 + i + 16) * ldw;
    const float* wr2   = W + (size_t)(n0 + i + 32) * ldw;
    const float* wr3   = W + (size_t)(n0 + i + 48) * ldw;

    v8f acc[2][4];
#pragma unroll
    for (int p = 0; p < 2; ++p)
#pragma unroll
        for (int j = 0; j < 4; ++j) acc[p][j] = (v8f){};

    FragNT cur, nxt;
    load_nt(cur, arow0, arow1, wr0, wr1, wr2, wr3, 4 * h);

#pragma unroll 2
    for (int kb = 0; kb < K - 8; kb += 8) {
        load_nt(nxt, arow0, arow1, wr0, wr1, wr2, wr3, kb + 8 + 4 * h);
        acc[0][0] = WMMA_F32(cur.a0.lo, cur.b0.lo, acc[0][0]);
        acc[0][1] = WMMA_F32(cur.a0.lo, cur.b1.lo, acc[0][1]);
        acc[0][2] = WMMA_F32(cur.a0.lo, cur.b2.lo, acc[0][2]);
        acc[0][3] = WMMA_F32(cur.a0.lo, cur.b3.lo, acc[0][3]);
        acc[1][0] = WMMA_F32(cur.a1.lo, cur.b0.lo, acc[1][0]);
        acc[1][1] = WMMA_F32(cur.a1.lo, cur.b1.lo, acc[1][1]);
        acc[1][2] = WMMA_F32(cur.a1.lo, cur.b2.lo, acc[1][2]);
        acc[1][3] = WMMA_F32(cur.a1.lo, cur.b3.lo, acc[1][3]);
        acc[0][0] = WMMA_F32(cur.a0.hi, cur.b0.hi, acc[0][0]);
        acc[0][1] = WMMA_F32(cur.a0.hi, cur.b1.hi, acc[0][1]);
        acc[0][2] = WMMA_F32(cur.a0.hi, cur.b2.hi, acc[0][2]);
        acc[0][3] = WMMA_F32(cur.a0.hi, cur.b3.hi, acc[0][3]);
        acc[1][0] = WMMA_F32(cur.a1.hi, cur.b0.hi, acc[1][0]);
        acc[1][1] = WMMA_F32(cur.a1.hi, cur.b1.hi, acc[1][1]);
        acc[1][2] = WMMA_F32(cur.a1.hi, cur.b2.hi, acc[1][2]);
        acc[1][3] = WMMA_F32(cur.a1.hi, cur.b3.hi, acc[1][3]);
        cur = nxt;
    }
    // epilogue chunk
    acc[0][0] = WMMA_F32(cur.a0.lo, cur.b0.lo, acc[0][0]);
    acc[0][1] = WMMA_F32(cur.a0.lo, cur.b1.lo, acc[0][1]);
    acc[0][2] = WMMA_F32(cur.a0.lo, cur.b2.lo, acc[0][2]);
    acc[0][3] = WMMA_F32(cur.a0.lo, cur.b3.lo, acc[0][3]);
    acc[1][0] = WMMA_F32(cur.a1.lo, cur.b0.lo, acc[1][0]);
    acc[1][1] = WMMA_F32(cur.a1.lo, cur.b1.lo, acc[1][1]);
    acc[1][2] = WMMA_F32(cur.a1.lo, cur.b2.lo, acc[1][2]);
    acc[1][3] = WMMA_F32(cur.a1.lo, cur.b3.lo, acc[1][3]);
    acc[0][0] = WMMA_F32(cur.a0.hi, cur.b0.hi, acc[0][0]);
    acc[0][1] = WMMA_F32(cur.a0.hi, cur.b1.hi, acc[0][1]);
    acc[0][2] = WMMA_F32(cur.a0.hi, cur.b2.hi, acc[0][2]);
    acc[0][3] = WMMA_F32(cur.a0.hi, cur.b3.hi, acc[0][3]);
    acc[1][0] = WMMA_F32(cur.a1.hi, cur.b0.hi, acc[1][0]);
    acc[1][1] = WMMA_F32(cur.a1.hi, cur.b1.hi, acc[1][1]);
    acc[1][2] = WMMA_F32(cur.a1.hi, cur.b2.hi, acc[1][2]);
    acc[1][3] = WMMA_F32(cur.a1.hi, cur.b3.hi, acc[1][3]);

    const int col = n0 + i;
    float bv[4] = {0.f, 0.f, 0.f, 0.f};
    if (bias) {
        bv[0] = bias[col];      bv[1] = bias[col + 16];
        bv[2] = bias[col + 32]; bv[3] = bias[col + 48];
    }
#pragma unroll
    for (int p = 0; p < 2; ++p) {
#pragma unroll
        for (int v = 0; v < 8; ++v) {
            float* crow = C + (size_t)(m0 + p * 16 + v + 8 * h) * ldc + col;
            crow[ 0] = acc[p][0][v] + bv[0];
            crow[16] = acc[p][1][v] + bv[1];
            crow[32] = acc[p][2][v] + bv[2];
            crow[48] = acc[p][3][v] + bv[3];
        }
    }
}

// ---------------------------------------------------------------------------
// Batched NN GEMM via WMMA f32:  C[z] = A[z][M,K] * Bm[z][K,N]
// Wave: 32 x 64, K-chunk 8, double-buffered. B gathered column-wise.
// grid = (M/32, N/256, batch)
// ---------------------------------------------------------------------------
struct FragNN { v4f a0, a1; v2f b[2][4]; };

__device__ __forceinline__ void load_nn(FragNN& f,
    const float* arow0, const float* arow1, const float* bbas,
    int kb, int h, int n, int ldb)
{
    f.a0 = *(const v4f*)(arow0 + kb + 4 * h);
    f.a1 = *(const v4f*)(arow1 + kb + 4 * h);
#pragma unroll
    for (int s = 0; s < 2; ++s) {                     // k-step within chunk
        const float* br0 = bbas + (size_t)(kb + 4 * s + 2 * h) * ldb;
        const float* br1 = br0 + ldb;
#pragma unroll
        for (int j = 0; j < 4; ++j) {
            f.b[s][j].x = br0[n + 16 * j];
            f.b[s][j].y = br1[n + 16 * j];
        }
    }
}

__global__ __launch_bounds__(128) void gemm_nn_wmma_batched(
    const float* __restrict__ A, const float* __restrict__ Bm, float* __restrict__ C,
    int K, int lda, int ldb, int ldc,
    long strideA, long strideB, long strideC)
{
    const int lane = threadIdx.x & 31;
    const int wave = threadIdx.x >> 5;
    const int m0   = blockIdx.x * 32;
    const int n0   = blockIdx.y * 256 + wave * 64;
    const int i    = lane & 15;
    const int h    = lane >> 4;
    const int z    = blockIdx.z;

    const float* arow0 = A + z * strideA + (size_t)(m0 + i) * lda;
    const float* arow1 = arow0 + (size_t)16 * lda;
    const float* bbas  = Bm + z * strideB;
    float*       cbas  = C + z * strideC;
    const int n = n0 + i;

    // NOTE: chunk's slot->k permutation for A is {kb+4h..kb+4h+3}; B's gathered
    // rows use a different one (s selects lo/hi step, rows kb+4s+2h, +1; A's step s holds kb+4h+2s, +1).
    v8f acc[2][4];
#pragma unroll
    for (int p = 0; p < 2; ++p)
#pragma unroll
        for (int j = 0; j < 4; ++j) acc[p][j] = (v8f){};

    FragNN cur, nxt;
    load_nn(cur, arow0, arow1, bbas, 0, h, n, ldb);

#pragma unroll 2
    for (int kb = 0; kb < K - 8; kb += 8) {
        load_nn(nxt, arow0, arow1, bbas, kb + 8, h, n, ldb);
        // A.lo pairs with B rows kb+0+2h / +1  <-> s=0 ; A.hi with kb+4+2h <-> s=1
        acc[0][0] = WMMA_F32(cur.a0.lo, cur.b[0][0], acc[0][0]);
        acc[0][1] = WMMA_F32(cur.a0.lo, cur.b[0][1], acc[0][1]);
        acc[0][2] = WMMA_F32(cur.a0.lo, cur.b[0][2], acc[0][2]);
        acc[0][3] = WMMA_F32(cur.a0.lo, cur.b[0][3], acc[0][3]);
        acc[1][0] = WMMA_F32(cur.a1.lo, cur.b[0][0], acc[1][0]);
        acc[1][1] = WMMA_F32(cur.a1.lo, cur.b[0][1], acc[1][1]);
        acc[1][2] = WMMA_F32(cur.a1.lo, cur.b[0][2], acc[1][2]);
        acc[1][3] = WMMA_F32(cur.a1.lo, cur.b[0][3], acc[1][3]);
        acc[0][0] = WMMA_F32(cur.a0.hi, cur.b[1][0], acc[0][0]);
        acc[0][1] = WMMA_F32(cur.a0.hi, cur.b[1][1], acc[0][1]);
        acc[0][2] = WMMA_F32(cur.a0.hi, cur.b[1][2], acc[0][2]);
        acc[0][3] = WMMA_F32(cur.a0.hi, cur.b[1][3], acc[0][3]);
        acc[1][0] = WMMA_F32(cur.a1.hi, cur.b[1][0], acc[1][0]);
        acc[1][1] = WMMA_F32(cur.a1.hi, cur.b[1][1], acc[1][1]);
        acc[1][2] = WMMA_F32(cur.a1.hi, cur.b[1][2], acc[1][2]);
        acc[1][3] = WMMA_F32(cur.a1.hi, cur.b[1][3], acc[1][3]);
        cur = nxt;
    }
    acc[0][0] = WMMA_F32(cur.a0.lo, cur.b[0][0], acc[0][0]);
    acc[0][1] = WMMA_F32(cur.a0.lo, cur.b[0][1], acc[0][1]);
    acc[0][2] = WMMA_F32(cur.a0.lo, cur.b[0][2], acc[0][2]);
    acc[0][3] = WMMA_F32(cur.a0.lo, cur.b[0][3], acc[0][3]);
    acc[1][0] = WMMA_F32(cur.a1.lo, cur.b[0][0], acc[1][0]);
    acc[1][1] = WMMA_F32(cur.a1.lo, cur.b[0][1], acc[1][1]);
    acc[1][2] = WMMA_F32(cur.a1.lo, cur.b[0][2], acc[1][2]);
    acc[1][3] = WMMA_F32(cur.a1.lo, cur.b[0][3], acc[1][3]);
    acc[0][0] = WMMA_F32(cur.a0.hi, cur.b[1][0], acc[0][0]);
    acc[0][1] = WMMA_F32(cur.a0.hi, cur.b[1][1], acc[0][1]);
    acc[0][2] = WMMA_F32(cur.a0.hi, cur.b[1][2], acc[0][2]);
    acc[0][3] = WMMA_F32(cur.a0.hi, cur.b[1][3], acc[0][3]);
    acc[1][0] = WMMA_F32(cur.a1.hi, cur.b[1][0], acc[1][0]);
    acc[1][1] = WMMA_F32(cur.a1.hi, cur.b[1][1], acc[1][1]);
    acc[1][2] = WMMA_F32(cur.a1.hi, cur.b[1][2], acc[1][2]);
    acc[1][3] = WMMA_F32(cur.a1.hi, cur.b[1][3], acc[1][3]);

#pragma unroll
    for (int p = 0; p < 2; ++p) {
#pragma unroll
        for (int v = 0; v < 8; ++v) {
            float* crow = cbas + (size_t)(m0 + p * 16 + v + 8 * h) * ldc + n;
            crow[ 0] = acc[p][0][v];
            crow[16] = acc[p][1][v];
            crow[32] = acc[p][2][v];
            crow[48] = acc[p][3][v];
        }
    }
}

// ---------------------------------------------------------------------------
// Energy: e[b,q,k] = sum_d v[d]*tanh(qp[b,q,d] + kp[b,k,d]), d = 256.
// Block: 256 threads, tile = 16 q x 32 k, full d in LDS (49 KB).
// grid = (BATCH * NQ/16, NKV/32). Writes straight into the alpha region.
// ---------------------------------------------------------------------------
#define EK_Q 16
#define EK_K 32
__global__ __launch_bounds__(256) void energy_kernel(
    const float* __restrict__ qp, const float* __restrict__ kp,
    const float* __restrict__ vvec, float* __restrict__ energy)
{
    __shared__ float sq[EK_Q * ADIM];   // 16 KB
    __shared__ float sk[EK_K * ADIM];   // 32 KB
    __shared__ float sv[ADIM];          //  1 KB

    const int b  = blockIdx.x >> 3;
    const int q0 = (blockIdx.x & 7) * EK_Q;
    const int k0 = blockIdx.y * EK_K;
    const int tid = threadIdx.x;

    const float4* qsrc = (const float4*)(qp + ((size_t)b * NQ  + q0) * ADIM);
    const float4* ksrc = (const float4*)(kp + ((size_t)b * NKV + k0) * ADIM);

#pragma unroll
    for (int i = tid; i < EK_Q * ADIM / 4; i += 256) ((float4*)sq)[i] = qsrc[i];
#pragma unroll
    for (int i = tid; i < EK_K * ADIM / 4; i += 256) ((float4*)sk)[i] = ksrc[i];
    if (tid < ADIM / 4) ((float4*)sv)[tid] = ((const float4*)vvec)[tid];
    __syncthreads();

    const int q  = tid & 15;
    const int kb = (tid >> 4) * 2;            // each thread: 2 k's
    const float4* sq4  = (const float4*)(sq + q * ADIM);
    const float4* sk40 = (const float4*)(sk + (kb + 0) * ADIM);
    const float4* sk41 = (const float4*)(sk + (kb + 1) * ADIM);
    const float4* sv4  = (const float4*)sv;

    float acc0 = 0.f, acc1 = 0.f;
#pragma unroll 4
    for (int d = 0; d < ADIM / 4; ++d) {
        float4 qv = sq4[d];
        float4 vv = sv4[d];
        float4 ka = sk40[d];
        float4 kc = sk41[d];
        acc0 += vv.x * fast_tanh(qv.x + ka.x) + vv.y * fast_tanh(qv.y + ka.y)
              + vv.z * fast_tanh(qv.z + ka.z) + vv.w * fast_tanh(qv.w + ka.w);
        acc1 += vv.x * fast_tanh(qv.x + kc.x) + vv.y * fast_tanh(qv.y + kc.y)
              + vv.z * fast_tanh(qv.z + kc.z) + vv.w * fast_tanh(qv.w + kc.w);
    }
    float* erow = energy + ((size_t)b * NQ + q0 + q) * NKV + k0 + kb;
    erow[0] = acc0;
    erow[1] = acc1;
}

// ---------------------------------------------------------------------------
// Softmax in place over rows of 1024. block = 256 threads (8 waves), 1 row.
// ---------------------------------------------------------------------------
__global__ __launch_bounds__(256) void softmax_kernel(float* __restrict__ alpha)
{
    __shared__ float red[8];
    const int tid = threadIdx.x;
    float4* row = (float4*)(alpha + (size_t)blockIdx.x * NKV);

    float4 x = row[tid];
    float m = fmaxf(fmaxf(x.x, x.y), fmaxf(x.z, x.w));
#pragma unroll
    for (int off = 16; off > 0; off >>= 1) m = fmaxf(m, __shfl_xor(m, off, 32));
    if ((tid & 31) == 0) red[tid >> 5] = m;
    __syncthreads();
    float mall = red[0];
#pragma unroll
    for (int i = 1; i < 8; ++i) mall = fmaxf(mall, red[i]);
    __syncthreads();

    const float L2E = 1.4426950408889634f;
    float4 e;
    e.x = __builtin_amdgcn_exp2f((x.x - mall) * L2E);
    e.y = __builtin_amdgcn_exp2f((x.y - mall) * L2E);
    e.z = __builtin_amdgcn_exp2f((x.z - mall) * L2E);
    e.w = __builtin_amdgcn_exp2f((x.w - mall) * L2E);
    float s = e.x + e.y + e.z + e.w;
#pragma unroll
    for (int off = 16; off > 0; off >>= 1) s += __shfl_xor(s, off, 32);
    if ((tid & 31) == 0) red[tid >> 5] = s;
    __syncthreads();
    float sall = red[0];
#pragma unroll
    for (int i = 1; i < 8; ++i) sall += red[i];

    float inv = __builtin_amdgcn_rcpf(sall);
    e.x *= inv; e.y *= inv; e.z *= inv; e.w *= inv;
    row[tid] = e;
}

// ---------------------------------------------------------------------------
// Launch: q,kv,W_q,W_k,v,W_v,W_o,b_o -> (out, alpha)
// context = (alpha @ kv) @ W_v^T  (algebraic refactor of alpha @ (kv @ W_v^T))
// ---------------------------------------------------------------------------
extern "C" void kernel_launch(void* const* d_in, const int* in_sizes, int n_in,
                              void* d_out, int out_size, void* d_ws, size_t ws_size,
                              hipStream_t stream)
{
    (void)in_sizes; (void)n_in; (void)out_size; (void)ws_size;
    const float* q   = (const float*)d_in[0];   // [B,NQ,QDIM]
    const float* kv  = (const float*)d_in[1];   // [B,NKV,KVDIM]
    const float* W_q = (const float*)d_in[2];   // [ADIM,QDIM]
    const float* W_k = (const float*)d_in[3];   // [ADIM,KVDIM]
    const float* vv  = (const float*)d_in[4];   // [1,ADIM]
    const float* W_v = (const float*)d_in[5];   // [QDIM,KVDIM]
    const float* W_o = (const float*)d_in[6];   // [QDIM,QDIM]
    const float* b_o = (const float*)d_in[7];   // [QDIM]

    float* out   = (float*)d_out;                       // [B,NQ,QDIM] = 262144
    float* alpha = out + (size_t)BATCH * NQ * QDIM;     // [B,NQ,NKV]  = 524288

    float* ws  = (float*)d_ws;
    float* qp  = ws;                                    // [B*NQ,  ADIM]  131072
    float* kp  = qp + (size_t)BATCH * NQ  * ADIM;       // [B*NKV, ADIM] 1048576
    float* t   = kp + (size_t)BATCH * NKV * ADIM;       // [B*NQ, KVDIM]  393216
    float* ctx = t  + (size_t)BATCH * NQ  * KVDIM;      // [B*NQ, QDIM]   262144

    const int MQ = BATCH * NQ;    // 512
    const int MK = BATCH * NKV;   // 4096

    // 1) q_proj = q @ W_q^T : [512,512]x[256,512]^T -> [512,256]
    gemm_nt_wmma<<<dim3(MQ / 32, ADIM / 256), 128, 0, stream>>>(
        q, W_q, nullptr, qp, QDIM, QDIM, QDIM, ADIM);

    // 2) k_proj = kv @ W_k^T : [4096,768]x[256,768]^T -> [4096,256]
    gemm_nt_wmma<<<dim3(MK / 32, ADIM / 256), 128, 0, stream>>>(
        kv, W_k, nullptr, kp, KVDIM, KVDIM, KVDIM, ADIM);

    // 3) energy -> alpha region of d_out
    energy_kernel<<<dim3(BATCH * NQ / EK_Q, NKV / EK_K), 256, 0, stream>>>(
        qp, kp, vv, alpha);

    // 4) softmax rows of 1024, in place
    softmax_kernel<<<dim3(BATCH * NQ), 256, 0, stream>>>(alpha);

    // 5) t[b] = alpha[b] @ kv[b] : [128,1024]x[1024,768] -> [128,768]
    gemm_nn_wmma_batched<<<dim3(NQ / 32, KVDIM / 256, BATCH), 128, 0, stream>>>(
        alpha, kv, t, NKV, NKV, KVDIM, KVDIM,
        (long)NQ * NKV, (long)NKV * KVDIM, (long)NQ * KVDIM);

    // 6) context = t @ W_v^T : [512,768]x[512,768]^T -> [512,512]
    gemm_nt_wmma<<<dim3(MQ / 32, QDIM / 256), 128, 0, stream>>>(
        t, W_v, nullptr, ctx, KVDIM, KVDIM, KVDIM, QDIM);

    // 7) out = context @ W_o^T + b_o : [512,512]x[512,512]^T -> [512,512]
    gemm_nt_wmma<<<dim3(MQ / 32, QDIM / 256), 128, 0, stream>>>(
        ctx, W_o, b_o, out, QDIM, QDIM, QDIM, QDIM);
}